// autoencoder_73804718014470
// MI455X (gfx1250) — compile-verified
//
#include <hip/hip_runtime.h>
#include <hip/hip_bf16.h>
#include <math.h>

// ---------------------------------------------------------------------------
// Problem constants
// ---------------------------------------------------------------------------
#define BATCH     2048
#define AXIS_IN   3
#define AXIS_LAT  2
#define DIM_A     6144              // BATCH * AXIS_IN
#define DIM_D     4096              // BATCH * (AXIS_IN-1)
#define N2        ((size_t)BATCH * BATCH)

typedef __attribute__((ext_vector_type(2))) float v2f;
typedef __attribute__((ext_vector_type(8))) float v8f;

// ---------------------------------------------------------------------------
// Memory-bound matvec: y[row] = act(W[row,:] . x + b[row])
// One wave (32 lanes) per output row; x cached in LDS; float4 coalesced loads.
// mode 0 = ReLU, 1 = Swish (z * sigmoid(z))
// ---------------------------------------------------------------------------
__global__ __launch_bounds__(256)
void matvec_act_kernel(const float* __restrict__ W,
                       const float* __restrict__ bias,
                       const float* __restrict__ x,
                       float* __restrict__ y,
                       int fan_in, int fan_out, int mode) {
    __shared__ float xs[DIM_A];

    const int tid = threadIdx.x;
    const int n4  = fan_in >> 2;

    // cooperative load of x into LDS (float4 granularity)
    const float4* x4  = reinterpret_cast<const float4*>(x);
    float4*       xs4 = reinterpret_cast<float4*>(xs);
    for (int i = tid; i < n4; i += blockDim.x) xs4[i] = x4[i];
    __syncthreads();

    const int lane = tid & 31;
    const int wave = tid >> 5;
    const int row  = blockIdx.x * (blockDim.x >> 5) + wave;
    if (row >= fan_out) return;

    const float4* w4 = reinterpret_cast<const float4*>(W + (size_t)row * fan_in);
    float acc = 0.0f;
    #pragma unroll 4
    for (int i = lane; i < n4; i += 32) {
        float4 w = w4[i];
        float4 v = xs4[i];
        acc += w.x * v.x + w.y * v.y + w.z * v.z + w.w * v.w;
    }
    // wave32 butterfly reduction
    #pragma unroll
    for (int off = 16; off > 0; off >>= 1)
        acc += __shfl_xor(acc, off, 32);

    if (lane == 0) {
        float z = acc + bias[row];
        float r;
        if (mode == 0) {
            r = z > 0.0f ? z : 0.0f;                 // ReLU
        } else {
            r = z * (1.0f / (1.0f + __expf(-z)));    // Swish
        }
        y[row] = r;
    }
}

// ---------------------------------------------------------------------------
// Pairwise Euclidean distance matrix via V_WMMA_F32_16X16X4_F32.
//   d2[i][j] = |c_i|^2 + |c_j|^2 - 2 c_i . c_j
// A is pre-scaled by -2, C seeded with s_i + s_j, so one WMMA per 16x16 tile
// yields d2 directly. dist = d2 > 0 ? v_sqrt_f32(d2) : 0.
// One wave per 16x16 tile. D (2 or 3) is a compile-time constant so the norm
// loops fully unroll and K-padding guards constant-fold.
// ---------------------------------------------------------------------------
template <int D>
__global__ __launch_bounds__(128)
void pairwise_wmma_kernel(const float* __restrict__ cord,
                          float* __restrict__ out) {
    constexpr int B  = BATCH;
    const int lane = threadIdx.x & 31;
    const int wave = threadIdx.x >> 5;
    const int tile = blockIdx.x * (blockDim.x >> 5) + wave;
    const int nt   = B >> 4;            // tiles per dimension (128)
    const int ti   = tile / nt;
    const int tj   = tile - ti * nt;

    const int half = lane >> 4;         // 0: K=0,1   1: K=2,3
    const int l16  = lane & 15;
    const int k0   = half * 2;

    // A-matrix (16x4 f32): lane l16 -> row M, v0 = K=k0, v1 = K=k0+1
    const int rowA = ti * 16 + l16;
    v2f a;
    a.x = (k0     < D) ? -2.0f * cord[rowA * D + k0    ] : 0.0f;
    a.y = (k0 + 1 < D) ? -2.0f * cord[rowA * D + k0 + 1] : 0.0f;

    // B-matrix (4x16 f32): lane l16 -> col N, v0 = K=k0, v1 = K=k0+1
    const int colB = tj * 16 + l16;
    v2f bb;
    bb.x = (k0     < D) ? cord[colB * D + k0    ] : 0.0f;
    bb.y = (k0 + 1 < D) ? cord[colB * D + k0 + 1] : 0.0f;

    // squared norm of this lane's column point (fully unrolled)
    float sc = 0.0f;
    #pragma unroll
    for (int k = 0; k < D; ++k) { float v = cord[colB * D + k]; sc += v * v; }

    // C accumulator: C[r] corresponds to (M = ti*16 + r + 8*half, N = colB)
    v8f c;
    #pragma unroll
    for (int r = 0; r < 8; ++r) {
        const int rowM = ti * 16 + r + 8 * half;
        float sr = 0.0f;
        #pragma unroll
        for (int k = 0; k < D; ++k) { float v = cord[rowM * D + k]; sr += v * v; }
        c[r] = sr + sc;
    }

    // D = (-2A) x B + (s_i + s_j)  ->  squared distances
    c = __builtin_amdgcn_wmma_f32_16x16x4_f32(
            /*neg_a=*/false, a, /*neg_b=*/false, bb,
            /*c_mod=*/(short)0, c, /*reuse_a=*/false, /*reuse_b=*/false);

    #pragma unroll
    for (int r = 0; r < 8; ++r) {
        const int rowM = ti * 16 + r + 8 * half;
        float d2   = c[r];
        float dist = (d2 > 0.0f) ? __builtin_amdgcn_sqrtf(d2) : 0.0f;
        out[(size_t)rowM * B + colB] = dist;
    }
}

// ---------------------------------------------------------------------------
// Launch: 7 matvec layers (ping-pong in d_ws), then 2 WMMA pairwise kernels.
// Inputs: d_in[0]=x, d_in[1+2i]=W_{i+1}, d_in[2+2i]=b_{i+1}
// Output: [ in_dist (N2) | lat_dist (N2) | lat_repr (4096) ]
// ---------------------------------------------------------------------------
extern "C" void kernel_launch(void* const* d_in, const int* in_sizes, int n_in,
                              void* d_out, int out_size, void* d_ws, size_t ws_size,
                              hipStream_t stream) {
    const float* x = (const float*)d_in[0];
    const float* W[7];
    const float* b[7];
    for (int i = 0; i < 7; ++i) {
        W[i] = (const float*)d_in[1 + 2 * i];
        b[i] = (const float*)d_in[2 + 2 * i];
    }

    float* out = (float*)d_out;
    float* ws0 = (float*)d_ws;
    float* ws1 = ws0 + DIM_A;
    float* lat = out + 2 * N2;          // lat_repr lives in the output buffer

    const dim3 blk(256);
    // 8 rows per block (one wave each)
    matvec_act_kernel<<<DIM_A / 8, blk, 0, stream>>>(W[0], b[0], x,   ws0, DIM_A, DIM_A, 0);
    matvec_act_kernel<<<DIM_A / 8, blk, 0, stream>>>(W[1], b[1], ws0, ws1, DIM_A, DIM_A, 0);
    matvec_act_kernel<<<DIM_A / 8, blk, 0, stream>>>(W[2], b[2], ws1, ws0, DIM_A, DIM_A, 0);
    matvec_act_kernel<<<DIM_A / 8, blk, 0, stream>>>(W[3], b[3], ws0, ws1, DIM_A, DIM_A, 0);
    matvec_act_kernel<<<DIM_A / 8, blk, 0, stream>>>(W[4], b[4], ws1, ws0, DIM_A, DIM_A, 0);
    matvec_act_kernel<<<DIM_A / 8, blk, 0, stream>>>(W[5], b[5], ws0, ws1, DIM_A, DIM_A, 0);
    matvec_act_kernel<<<DIM_D / 8, blk, 0, stream>>>(W[6], b[6], ws1, lat, DIM_A, DIM_D, 1);

    // 128x128 tiles of 16x16 = 16384 waves; 4 waves (128 threads) per block
    const int ntiles = (BATCH / 16) * (BATCH / 16);
    pairwise_wmma_kernel<AXIS_IN ><<<ntiles / 4, dim3(128), 0, stream>>>(x,   out);
    pairwise_wmma_kernel<AXIS_LAT><<<ntiles / 4, dim3(128), 0, stream>>>(lat, out + N2);
}